// DistillationLoss_75436805587351
// MI455X (gfx1250) — compile-verified
//
#include <hip/hip_runtime.h>
#include <cstdint>

// ---------------- problem constants (from reference) ----------------
#define IGNORE_INDEX (-100)
#define BATCH     4
#define SEQ       512
#define V_S       32000          // student vocab
#define V_T       50257          // teacher vocab
#define NSORT_S   32768          // pow2 >= V_S (128 KB LDS)
#define NSORT_T   65536          // pow2 >= V_T (256 KB LDS -- needs CDNA5 320KB LDS)
#define NTHREADS  1024           // 32 waves (wave32)
#define NTASKS    (BATCH * SEQ)

typedef unsigned int v4u __attribute__((ext_vector_type(4)));
typedef int          v4i __attribute__((ext_vector_type(4)));
typedef int          v8i __attribute__((ext_vector_type(8)));
typedef float        v2f __attribute__((ext_vector_type(2)));
typedef float        v8f __attribute__((ext_vector_type(8)));

// ---------------- TDM: DMA one contiguous f32 row global -> LDS ----------------
// Descriptor per CDNA5 ISA ch.8 (group0: count/lds_addr/global_addr/type,
// group1: data_size=4B, tensor_dim0=n, tensor_dim1=1, tile=(n,1)).
// Must be reached by exactly one wave (EXEC is ignored; per-wave issue).
__device__ __forceinline__ void tdm_load_row(const float* gptr, float* lptr, int nelem) {
#if __has_builtin(__builtin_amdgcn_tensor_load_to_lds)
  unsigned long long ga = (unsigned long long)(uintptr_t)gptr;
  unsigned int lds = (unsigned int)(uintptr_t)lptr;   // flat shared addr low 32 = LDS byte addr

  v4u g0;
  g0[0] = 1u;                                         // count = 1 valid descriptor
  g0[1] = lds;                                        // lds_addr (bytes)
  g0[2] = (unsigned int)(ga & 0xFFFFFFFFu);           // global_addr[31:0]
  g0[3] = (unsigned int)((ga >> 32) & 0x01FFFFFFu)    // global_addr[56:32]
        | (2u << 30);                                 // type = 2 ("image")

  unsigned un = (unsigned)nelem;
  v8i g1;
  g1[0] = (int)(2u << 16);                            // wg_mask=0, data_size=2 (4 bytes)
  g1[1] = (int)((un & 0xFFFFu) << 16);                // tensor_dim0[15:0]
  g1[2] = (int)((un >> 16) | (1u << 16));             // tensor_dim0[31:16] | tensor_dim1=1
  g1[3] = (int)(un << 16);                            // tile_dim0 = n (fits 16 bits)
  g1[4] = 1;                                          // tile_dim1 = 1, tile_dim2 = 0
  g1[5] = (int)un;                                    // tensor_dim0_stride lo
  g1[6] = 0;
  g1[7] = 0;

  v4i z4 = {0, 0, 0, 0};
#if __clang_major__ >= 23
  v8i z8 = {0, 0, 0, 0, 0, 0, 0, 0};
  __builtin_amdgcn_tensor_load_to_lds(g0, g1, z4, z4, z8, 0);
#else
  __builtin_amdgcn_tensor_load_to_lds(g0, g1, z4, z4, 0);
#endif
#else
  // Fallback (should not trigger on gfx1250 toolchains): wave-0 lanes copy.
  for (int i = (int)(threadIdx.x & 31u); i < nelem; i += 32) lptr[i] = gptr[i];
#endif
}

__device__ __forceinline__ void tdm_wait0() {
#if __has_builtin(__builtin_amdgcn_s_wait_tensorcnt)
  __builtin_amdgcn_s_wait_tensorcnt(0);
#else
  asm volatile("s_wait_tensorcnt 0" ::: "memory");
#endif
}

// ---------------- reductions (wave32 shfl + WMMA f32 16x16x4 collapse) --------
__device__ __forceinline__ float blockReduceMax(float v, float* red) {
  for (int off = 16; off; off >>= 1) v = fmaxf(v, __shfl_xor(v, off));
  const int wave = threadIdx.x >> 5, lane = threadIdx.x & 31;
  __syncthreads();                       // protect red[] from previous use
  if (lane == 0) red[wave] = v;
  __syncthreads();
  if (wave == 0) {
    float t = red[lane];
    for (int off = 16; off; off >>= 1) t = fmaxf(t, __shfl_xor(t, off));
    if (lane == 0) red[0] = t;
  }
  __syncthreads();
  float r = red[0];
  __syncthreads();
  return r;
}

// Sum 32 per-wave partials (in red[0..31]) using V_WMMA_F32_16X16X4_F32:
// A(16x4) holds the 32 partials in rows 0..7, B = ones(4x16) -> D rows = row sums.
// Called by all 32 lanes of wave 0 (EXEC all ones, as WMMA requires).
__device__ __forceinline__ float wmma_sum32(const float* red, int lane) {
#if __has_builtin(__builtin_amdgcn_wmma_f32_16x16x4_f32)
  const int m  = lane & 15;
  const int hi = (lane >= 16) ? 2 : 0;   // lanes 16-31 hold K=2,3 (ISA A-layout 16x4)
  v2f a;
  if (m < 8) { a.x = red[4 * m + hi]; a.y = red[4 * m + hi + 1]; }
  else       { a.x = 0.0f;            a.y = 0.0f; }
  v2f ones; ones.x = 1.0f; ones.y = 1.0f;          // all-ones B under any layout
  v8f c = {};
  v8f d = __builtin_amdgcn_wmma_f32_16x16x4_f32(false, a, false, ones,
                                                (short)0, c, false, false);
  // Lanes 0-15 hold rows M=0..7 across the 8 D VGPRs; rows 8..15 are zero.
  return d[0] + d[1] + d[2] + d[3] + d[4] + d[5] + d[6] + d[7];
#else
  float t = red[lane];
  for (int off = 16; off; off >>= 1) t += __shfl_xor(t, off);
  return t;
#endif
}

__device__ __forceinline__ float blockReduceSum(float v, float* red) {
  for (int off = 16; off; off >>= 1) v += __shfl_xor(v, off);
  const int wave = threadIdx.x >> 5, lane = threadIdx.x & 31;
  __syncthreads();                       // protect red[] from previous use
  if (lane == 0) red[wave] = v;
  __syncthreads();
  if (wave == 0) {                       // wave-uniform branch: EXEC stays all-ones
    float t = wmma_sum32(red, lane);
    if (lane == 0) red[0] = t;
  }
  __syncthreads();
  float r = red[0];
  __syncthreads();
  return r;
}

// ---------------- softmax in place over LDS row ----------------
__device__ void softmax_inplace(float* buf, int V, float* red) {
  const int tid = threadIdx.x;
  float m = -3.402823466e38f;
  for (int i = tid; i < V; i += NTHREADS) m = fmaxf(m, buf[i]);
  m = blockReduceMax(m, red);
  float s = 0.0f;
  for (int i = tid; i < V; i += NTHREADS) {
    float e = __expf(buf[i] - m);
    buf[i] = e;
    s += e;
  }
  s = blockReduceSum(s, red);
  const float inv = 1.0f / s;
  for (int i = tid; i < V; i += NTHREADS) buf[i] *= inv;
  __syncthreads();
}

// ---------------- in-LDS bitonic sort, descending ----------------
__device__ void bitonic_desc(float* buf, unsigned n) {
  const unsigned tid = threadIdx.x;
  for (unsigned k = 2; k <= n; k <<= 1) {
    for (unsigned s = k >> 1; s > 0; s >>= 1) {
      for (unsigned p = tid; p < (n >> 1); p += NTHREADS) {
        unsigned i = ((p & ~(s - 1u)) << 1) | (p & (s - 1u));
        unsigned j = i + s;
        float a = buf[i], b = buf[j];
        bool sw = ((i & k) == 0u) ? (a < b) : (a > b);   // descending overall
        if (sw) { buf[i] = b; buf[j] = a; }
      }
      __syncthreads();
    }
  }
}

// ---------------- kernel 1: targets scan (idx/size/min_size), zero accum -------
// ws int header: [0..3]=s_idx [4..7]=s_size [8..11]=t_idx [12..15]=t_size [16..19]=min
__global__ void prep_kernel(const int* __restrict__ st, const int* __restrict__ tt,
                            int* __restrict__ hdrI, float* __restrict__ accum) {
  const int tid = threadIdx.x;
  if (tid < 8) {
    const int* tgt = (tid < 4) ? st : tt;
    const int  b   = tid & 3;
    const int* row = tgt + b * SEQ;
    int size = 0, lead = 0;
    bool leading = true;
    for (int i = 0; i < SEQ; ++i) {
      int  v   = row[i];
      bool ign = (v == IGNORE_INDEX);
      if (!ign) size++;
      if (leading) { if (ign) lead++; else leading = false; }
    }
    int idx = (row[0] == IGNORE_INDEX) ? (lead - 1) : 0;
    if (tid < 4) { hdrI[0 + b] = idx; hdrI[4 + b]  = size; }
    else         { hdrI[8 + b] = idx; hdrI[12 + b] = size; }
  }
  __syncthreads();
  if (tid < 4) {
    int a = hdrI[4 + tid], bsz = hdrI[12 + tid];
    hdrI[16 + tid] = a < bsz ? a : bsz;
    accum[tid] = 0.0f;
  }
}

// ---------------- kernel 2: per row-pair softmax + sort + L1 ----------------
__global__ __launch_bounds__(NTHREADS) void kd_kernel(
    const float* __restrict__ slog, const float* __restrict__ tlog,
    const int* __restrict__ hdrI, float* __restrict__ accum,
    float* __restrict__ scratch) {
  extern __shared__ float buf[];        // 65536 floats = 256 KB dynamic LDS
  __shared__ float red[32];
  const int tid = threadIdx.x;
  float* mySc = scratch + (size_t)blockIdx.x * NSORT_S;   // 128 KB per persistent WG

  for (int task = blockIdx.x; task < NTASKS; task += (int)gridDim.x) {
    const int b = task >> 9;            // SEQ = 512
    const int j = task & (SEQ - 1);
    if (j >= hdrI[16 + b]) continue;    // uniform across block

    // ---------------- student row ----------------
    {
      int pos = hdrI[b] + j;
      pos = pos < 0 ? 0 : (pos > SEQ - 1 ? SEQ - 1 : pos);
      const float* row = slog + (size_t)(b * SEQ + pos) * V_S;
      if ((tid >> 5) == 0) tdm_load_row(row, buf, V_S);        // wave 0 issues TDM
      for (int i = V_S + tid; i < NSORT_S; i += NTHREADS) buf[i] = 0.0f;
      if ((tid >> 5) == 0) tdm_wait0();                        // s_wait_tensorcnt 0
      __syncthreads();
      softmax_inplace(buf, V_S, red);
      bitonic_desc(buf, NSORT_S);
      // spill sorted student to per-WG global scratch (L2-resident)
      float4* s4 = (float4*)mySc;
      const float4* b4 = (const float4*)buf;
      for (int i = tid; i < NSORT_S / 4; i += NTHREADS) s4[i] = b4[i];
      __threadfence();
      __syncthreads();
    }

    // ---------------- teacher row + L1 ----------------
    float l1;
    {
      int pos = hdrI[8 + b] + j;
      pos = pos < 0 ? 0 : (pos > SEQ - 1 ? SEQ - 1 : pos);
      const float* row = tlog + (size_t)(b * SEQ + pos) * V_T;
      if ((tid >> 5) == 0) tdm_load_row(row, buf, V_T);
      for (int i = V_T + tid; i < NSORT_T; i += NTHREADS) buf[i] = 0.0f;
      if ((tid >> 5) == 0) tdm_wait0();
      __syncthreads();
      softmax_inplace(buf, V_T, red);
      bitonic_desc(buf, NSORT_T);
      float acc = 0.0f;
      for (int k = tid; k < NSORT_T; k += NTHREADS) {
        float sv = (k < NSORT_S) ? mySc[k] : 0.0f;   // student zero-padded above 32768
        acc += fabsf(buf[k] - sv);
      }
      l1 = blockReduceSum(acc, red);
    }
    if (tid == 0) atomicAdd(&accum[b], l1);          // global_atomic_add_f32
    __syncthreads();                                 // buf reuse fence before next task
  }
}

// ---------------- kernel 3: epilogue ----------------
__global__ void fin_kernel(const int* __restrict__ hdrI, const float* __restrict__ accum,
                           const float* __restrict__ loss, float* __restrict__ out) {
  if (threadIdx.x == 0) {
    float kd = 0.0f;
    for (int b = 0; b < BATCH; ++b) kd += accum[b] / (float)hdrI[16 + b];
    kd *= (1.0f / (float)BATCH);       // mean over batch; KD_WEIGHT = 1
    float ce = loss[0];                // CE_WEIGHT = 1
    out[0] = ce + kd;
    out[1] = ce;
    out[2] = kd;
  }
}

// ---------------- host launcher ----------------
extern "C" void kernel_launch(void* const* d_in, const int* in_sizes, int n_in,
                              void* d_out, int out_size, void* d_ws, size_t ws_size,
                              hipStream_t stream) {
  (void)in_sizes; (void)n_in; (void)out_size;
  const float* slog = (const float*)d_in[0];
  const float* tlog = (const float*)d_in[1];
  const int*   st   = (const int*)d_in[2];
  const int*   tt   = (const int*)d_in[3];
  const float* loss = (const float*)d_in[4];
  float* out = (float*)d_out;

  int*   hdrI    = (int*)d_ws;                       // 20 ints
  float* accum   = (float*)((char*)d_ws + 80);       // 4 floats
  float* scratch = (float*)((char*)d_ws + 256);      // per-WG sorted-student slices

  const size_t per_wg = (size_t)NSORT_S * sizeof(float);   // 128 KB
  int nwg = 1;
  if (ws_size > 256 + per_wg) {
    size_t cap = (ws_size - 256) / per_wg;
    nwg = (int)(cap < 512 ? cap : 512);              // persistent WGs, 1 per WGP (LDS-bound)
    if (nwg < 1) nwg = 1;
  }

  prep_kernel<<<1, 128, 0, stream>>>(st, tt, hdrI, accum);
  kd_kernel<<<nwg, NTHREADS, NSORT_T * sizeof(float), stream>>>(slog, tlog, hdrI, accum, scratch);
  fin_kernel<<<1, 32, 0, stream>>>(hdrI, accum, loss, out);
}